// GraphConv_27273042330337
// MI455X (gfx1250) — compile-verified
//
#include <hip/hip_runtime.h>

typedef float v2f __attribute__((ext_vector_type(2)));
typedef float v8f __attribute__((ext_vector_type(8)));

#define CH 64  // in_ch == out_ch == 64

// ---------------------------------------------------------------------------
// Kernel 1: dual GEMM via V_WMMA_F32_16X16X4_F32 (wave32, 16x16 tiles)
//   yself  = x @ W_self  + b_self   -> d_out
//   yneigh = x @ W_neigh + b_neigh  -> d_ws
// Each wave: one 16-row x 16-col output tile, BOTH weight matrices.
// Block = 256 threads = 8 waves = 2 row-tiles x 4 col-tiles.
// ---------------------------------------------------------------------------
__global__ __launch_bounds__(256) void graphconv_gemm_dual(
    const float* __restrict__ x,
    const float* __restrict__ Wself, const float* __restrict__ bself,
    const float* __restrict__ Wneigh, const float* __restrict__ bneigh,
    float* __restrict__ yself, float* __restrict__ yneigh,
    int rowTiles)
{
    const int wave = threadIdx.x >> 5;      // 0..7
    const int lane = threadIdx.x & 31;
    const int half = lane >> 4;             // 0: lanes 0-15, 1: lanes 16-31
    const int l    = lane & 15;

    const int rowtile = blockIdx.x * 2 + (wave >> 2);   // wave-uniform
    if (rowtile >= rowTiles) return;                    // uniform exit: EXEC stays all-1s
    const int coltile = wave & 3;

    const int m0 = rowtile * 16;
    const int n0 = coltile * 16;
    const int n  = n0 + l;                  // output column owned by this lane

    // C/D 16x16 f32 layout: VGPR v holds M = v + 8*half, N = l.
    // Bias depends only on N -> broadcast across all 8 accumulator VGPRs.
    v8f cs, cn;
    {
        const float bs = bself[n];
        const float bn = bneigh[n];
        #pragma unroll
        for (int v = 0; v < 8; ++v) { cs[v] = bs; cn[v] = bn; }
    }

    // A 16x4 f32 layout: lane l (half h) holds row M=m0+l, K = k0+2h, k0+2h+1
    const float* arow = x + (size_t)(m0 + l) * CH + 2 * half;

    #pragma unroll
    for (int k0 = 0; k0 < CH; k0 += 4) {
        const v2f a = *(const v2f*)(arow + k0);   // 8B-aligned contiguous K pair

        // B 4x16 f32 layout (mirror of A): lane l (half h) holds col N=n,
        // rows K = k0+2h, k0+2h+1 (stride CH apart in row-major W[in][out]).
        const int kr = k0 + 2 * half;
        v2f bsv, bnv;
        bsv.x = Wself [(size_t)kr       * CH + n];
        bsv.y = Wself [(size_t)(kr + 1) * CH + n];
        bnv.x = Wneigh[(size_t)kr       * CH + n];
        bnv.y = Wneigh[(size_t)(kr + 1) * CH + n];

        cs = __builtin_amdgcn_wmma_f32_16x16x4_f32(
                false, a, false, bsv, (short)0, cs, false, false);
        cn = __builtin_amdgcn_wmma_f32_16x16x4_f32(
                false, a, false, bnv, (short)0, cn, false, false);
    }

    // Write back: VGPR v -> row m0 + v + 8*half, column n.
    float* ps = yself  + n;
    float* pn = yneigh + n;
    #pragma unroll
    for (int v = 0; v < 8; ++v) {
        const size_t m = (size_t)(m0 + v + 8 * half);
        ps[m * CH] = cs[v];
        pn[m * CH] = cn[v];
    }
}

// ---------------------------------------------------------------------------
// Kernel 2: edge scatter. One wave per edge; each lane moves 2 channels.
//   out[col[e], :] += yneigh[row[e], :]
// unsafeAtomicAdd lowers to global_atomic_add_f32 (no-return).
// ---------------------------------------------------------------------------
__global__ __launch_bounds__(256) void graphconv_scatter(
    const int* __restrict__ edge_index,  // [2, E]
    const float* __restrict__ msg,       // [N, 64]
    float* __restrict__ out,             // [N, 64]
    int E)
{
    const int gid  = blockIdx.x * blockDim.x + threadIdx.x;
    const int e    = gid >> 5;
    if (e >= E) return;
    const int lane = gid & 31;

    const int r = edge_index[e];
    const int c = edge_index[E + e];

    const float2 v = *(const float2*)(msg + (size_t)r * CH + 2 * lane);
    float* p = out + (size_t)c * CH + 2 * lane;
    unsafeAtomicAdd(p,     v.x);
    unsafeAtomicAdd(p + 1, v.y);
}

// ---------------------------------------------------------------------------
// Kernel 3: in-place ReLU, float4-vectorized (N*64 is a multiple of 4).
// ---------------------------------------------------------------------------
__global__ __launch_bounds__(256) void graphconv_relu(float* __restrict__ out, int n4)
{
    const int i = blockIdx.x * blockDim.x + threadIdx.x;
    if (i >= n4) return;
    float4 v = ((const float4*)out)[i];
    v.x = fmaxf(v.x, 0.0f);
    v.y = fmaxf(v.y, 0.0f);
    v.z = fmaxf(v.z, 0.0f);
    v.w = fmaxf(v.w, 0.0f);
    ((float4*)out)[i] = v;
}

// ---------------------------------------------------------------------------
// Launch
// ---------------------------------------------------------------------------
extern "C" void kernel_launch(void* const* d_in, const int* in_sizes, int n_in,
                              void* d_out, int out_size, void* d_ws, size_t ws_size,
                              hipStream_t stream) {
    const float* x      = (const float*)d_in[0];
    const int*   ei     = (const int*)  d_in[1];
    const float* Wself  = (const float*)d_in[2];
    const float* bself  = (const float*)d_in[3];
    const float* Wneigh = (const float*)d_in[4];
    const float* bneigh = (const float*)d_in[5];

    float* out = (float*)d_out;
    float* ws  = (float*)d_ws;     // y_neigh staging: N*64 floats

    const int nNodes   = in_sizes[0] / CH;     // 50000
    const int E        = in_sizes[1] / 2;      // 800000
    const int rowTiles = (nNodes + 15) / 16;   // 3125

    // 1) dual GEMM (2 row-tiles per 256-thread block)
    const int gemmBlocks = (rowTiles + 1) / 2;
    graphconv_gemm_dual<<<gemmBlocks, 256, 0, stream>>>(
        x, Wself, bself, Wneigh, bneigh, out, ws, rowTiles);

    // 2) scatter-add of neighbor messages (1 wave per edge)
    const long long scatterThreads = (long long)E * 32;
    const int scatterBlocks = (int)((scatterThreads + 255) / 256);
    graphconv_scatter<<<scatterBlocks, 256, 0, stream>>>(ei, ws, out, E);

    // 3) ReLU in place
    const int n4 = nNodes * CH / 4;
    graphconv_relu<<<(n4 + 255) / 256, 256, 0, stream>>>(out, n4);
}